// TransformerBlockQuantum_65481071401500
// MI455X (gfx1250) — compile-verified
//
#include <hip/hip_runtime.h>

// ---------------------------------------------------------------------------
// Quantum transformer block for MI455X (gfx1250), wave32 + V_WMMA_F32_16X16X4.
// Quantum circuits reduced analytically to cos prefix-products (see analysis).
// ---------------------------------------------------------------------------

typedef float v2f __attribute__((ext_vector_type(2)));
typedef float v8f __attribute__((ext_vector_type(8)));

#define BB 8
#define SS 1024
#define EE 64
#define HH 8
#define DKK 8
#define MTOT (BB * SS)          // 8192 rows
#define FFN_N 256
#define LN_EPS 1e-5f

static __device__ __forceinline__ v8f wmma4(v2f a, v2f b, v8f c) {
    // D(16x16,f32) = A(16x4,f32) x B(4x16,f32) + C
    return __builtin_amdgcn_wmma_f32_16x16x4_f32(
        false, a, false, b, (short)0, c, false, false);
}

static __device__ __forceinline__ v8f zero8() {
    v8f z = {0.f, 0.f, 0.f, 0.f, 0.f, 0.f, 0.f, 0.f};
    return z;
}

// ===========================================================================
// Kernel 1: QKV projection (WMMA f32 GEMM 8192x64x64) fused with quantum map.
// grid = (512 m-tiles, 3 {q,k,v}); block = 128 (4 waves, one 16-col tile each)
// Output layout: [B, H, S, DK] fp32.
// ===========================================================================
__global__ void k_proj_quantum(const float* __restrict__ x,
                               const float* __restrict__ wq,
                               const float* __restrict__ wk,
                               const float* __restrict__ wv,
                               const float* __restrict__ head_rx,
                               float* __restrict__ qp,
                               float* __restrict__ kp,
                               float* __restrict__ vp) {
    __shared__ float tile[4][16][17];
    const int wave = threadIdx.x >> 5;
    const int lane = threadIdx.x & 31;
    const int ln = lane & 15;
    const int lh = lane >> 4;              // which half of the wave
    const int mtile = blockIdx.x;
    const int which = blockIdx.y;

    const float* w = (which == 0) ? wq : ((which == 1) ? wk : wv);
    float* outp = (which == 0) ? qp : ((which == 1) ? kp : vp);

    const int mrow = mtile * 16 + ln;      // A row (M = lane%16)
    const int nrow = wave * 16 + ln;       // B col -> weight row
    const int khalf = lh * 2;              // VGPR K pair base per ISA layout

    v8f acc = zero8();
#pragma unroll
    for (int s = 0; s < 16; ++s) {
        const int k = s * 4 + khalf;
        v2f a = { x[mrow * EE + k], x[mrow * EE + k + 1] };
        v2f b = { w[nrow * EE + k], w[nrow * EE + k + 1] };
        acc = wmma4(a, b, acc);
    }

    // D layout: VGPR i -> row (i + 8*lh), col = ln. Stash tile in LDS.
#pragma unroll
    for (int i = 0; i < 8; ++i)
        tile[wave][i + 8 * lh][ln] = acc[i];
    __syncthreads();

    // Quantum map: lane covers (row = ln, local head = lh).  Cols of this
    // 16-wide tile are 2 heads x 8 wires.  out_w = prod_{j<=w} cos(theta_j),
    // out_0 = prod_{j=1..7} cos(theta_j).
    const int h = (wave * 16 + lh * 8) >> 3;
    const int m = mtile * 16 + ln;
    const int bidx = m >> 10;
    const int sidx = m & 1023;

    float c[8];
#pragma unroll
    for (int j = 0; j < 8; ++j) {
        const float theta = tile[wave][ln][lh * 8 + j] + head_rx[h * DKK + j];
        c[j] = __cosf(theta);
    }
    float* dst = outp + (((bidx * HH + h) * SS) + sidx) * DKK;
    float out0 = 1.f;
#pragma unroll
    for (int j = 1; j < 8; ++j) out0 *= c[j];
    dst[0] = out0;
    float pref = c[0];
#pragma unroll
    for (int j = 1; j < 8; ++j) { pref *= c[j]; dst[j] = pref; }
}

// ===========================================================================
// Kernel 2: flash attention, DK=8, S=1024.  One wave per 16-query tile.
// grid = (16, B*H=64); block = 128 (4 waves -> 4 query tiles).
// scores = QK^T/sqrt(8) via WMMA, online softmax via shfl_xor(width 16),
// P x V via WMMA after an LDS layout bounce (D-layout -> A-layout).
// Writes ao in [B, S, H*DK] fp32.
// ===========================================================================
__global__ void k_flash_attn(const float* __restrict__ qp,
                             const float* __restrict__ kp,
                             const float* __restrict__ vp,
                             float* __restrict__ ao) {
    __shared__ float pbuf[4][16][17];
    const int wave = threadIdx.x >> 5;
    const int lane = threadIdx.x & 31;
    const int ln = lane & 15;
    const int lh = lane >> 4;
    const int bh = blockIdx.y;
    const int b = bh >> 3;
    const int h = bh & 7;
    const int qtile = blockIdx.x * 4 + wave;
    const int base = bh * SS * DKK;
    const float scale = 0.3535533906f;      // 1/sqrt(8)

    // Q fragments (A: 16x4 per step, 2 steps cover K=8)
    v2f qa[2];
#pragma unroll
    for (int s = 0; s < 2; ++s) {
        const int k = s * 4 + lh * 2;
        const float* qrow = qp + base + (qtile * 16 + ln) * DKK + k;
        qa[s] = { qrow[0], qrow[1] };
    }

    float mrow[8], lrow[8];
#pragma unroll
    for (int i = 0; i < 8; ++i) { mrow[i] = -3.0e38f; lrow[i] = 0.f; }
    v8f o = zero8();

    for (int kt = 0; kt < SS / 16; ++kt) {
        const int kbase = base + kt * 16 * DKK;

        // scores tile: 16q x 16k
        v8f sc = zero8();
#pragma unroll
        for (int s = 0; s < 2; ++s) {
            const int k = s * 4 + lh * 2;
            const float* krow = kp + kbase + ln * DKK + k;   // B[k][n]=K[n][k]
            v2f kb = { krow[0], krow[1] };
            sc = wmma4(qa[s], kb, sc);
        }

        // online softmax per row (row i lives in VGPR i, cols across 16 lanes)
#pragma unroll
        for (int i = 0; i < 8; ++i) {
            float sv = sc[i] * scale;
            float rmax = sv;
#pragma unroll
            for (int off = 8; off >= 1; off >>= 1)
                rmax = fmaxf(rmax, __shfl_xor(rmax, off, 16));
            const float mn = fmaxf(mrow[i], rmax);
            const float p = __expf(sv - mn);
            float rsum = p;
#pragma unroll
            for (int off = 8; off >= 1; off >>= 1)
                rsum += __shfl_xor(rsum, off, 16);
            const float esc = __expf(mrow[i] - mn);
            lrow[i] = lrow[i] * esc + rsum;
            o[i] *= esc;
            mrow[i] = mn;
            pbuf[wave][i + 8 * lh][ln] = p;    // D-layout -> LDS
        }
        __syncthreads();

        // P(16x16) x V(16x8 padded to 16): K=16 -> 4 WMMA steps
#pragma unroll
        for (int s = 0; s < 4; ++s) {
            const int k0 = s * 4 + lh * 2;
            v2f pa = { pbuf[wave][ln][k0], pbuf[wave][ln][k0 + 1] };
            const float bv0 = (ln < 8) ? vp[kbase + k0 * DKK + ln] : 0.f;
            const float bv1 = (ln < 8) ? vp[kbase + (k0 + 1) * DKK + ln] : 0.f;
            v2f vb = { bv0, bv1 };
            o = wmma4(pa, vb, o);
        }
        __syncthreads();   // protect pbuf (WAR) for next key tile
    }

    // finalize: divide by row sum, scatter to ao[B,S,H*DK]
#pragma unroll
    for (int i = 0; i < 8; ++i) {
        const int row = qtile * 16 + i + 8 * lh;
        if (ln < 8)
            ao[(b * SS + row) * EE + h * DKK + ln] = o[i] / lrow[i];
    }
}

// ===========================================================================
// Kernel 3: out-projection + residual + LayerNorm1 -> x1.
// grid = 512; block = 128 (4 waves, one 16-col tile each).
// ===========================================================================
__global__ void k_outproj_ln1(const float* __restrict__ ao,
                              const float* __restrict__ wo,
                              const float* __restrict__ x,
                              const float* __restrict__ g,
                              const float* __restrict__ bt,
                              float* __restrict__ x1) {
    __shared__ float tile[16][68];
    const int wave = threadIdx.x >> 5;
    const int lane = threadIdx.x & 31;
    const int ln = lane & 15;
    const int lh = lane >> 4;
    const int mtile = blockIdx.x;

    const int mrow = mtile * 16 + ln;
    const int nrow = wave * 16 + ln;

    v8f acc = zero8();
#pragma unroll
    for (int s = 0; s < 16; ++s) {
        const int k = s * 4 + lh * 2;
        v2f a = { ao[mrow * EE + k], ao[mrow * EE + k + 1] };
        v2f b = { wo[nrow * EE + k], wo[nrow * EE + k + 1] };
        acc = wmma4(a, b, acc);
    }
#pragma unroll
    for (int i = 0; i < 8; ++i)
        tile[i + 8 * lh][wave * 16 + ln] = acc[i];
    __syncthreads();

    if (threadIdx.x < 16) {
        const int r = threadIdx.x;
        const int m = mtile * 16 + r;
        float mean = 0.f;
        for (int c = 0; c < EE; ++c)
            mean += tile[r][c] + x[m * EE + c];
        mean *= (1.f / EE);
        float var = 0.f;
        for (int c = 0; c < EE; ++c) {
            const float d = tile[r][c] + x[m * EE + c] - mean;
            var += d * d;
        }
        var *= (1.f / EE);
        const float inv = __frsqrt_rn(var + LN_EPS);
        for (int c = 0; c < EE; ++c) {
            const float v = tile[r][c] + x[m * EE + c];
            x1[m * EE + c] = (v - mean) * inv * g[c] + bt[c];
        }
    }
}

// ===========================================================================
// Kernel 4: quantum FFN fused:
//   z = quantum(x1[:, :8])  -> relu -> @w1^T + b1 -> @w2^T + b2
//   out = LayerNorm2(x1 + ffn_out)
// grid = 512 (16-token tiles); block = 128.
// ===========================================================================
__global__ void k_ffn_ln2(const float* __restrict__ x1,
                          const float* __restrict__ ffn_ry,
                          const float* __restrict__ w1,
                          const float* __restrict__ b1,
                          const float* __restrict__ w2,
                          const float* __restrict__ b2,
                          const float* __restrict__ g,
                          const float* __restrict__ bt,
                          float* __restrict__ out) {
    __shared__ float zb[16][12];        // relu(z): 16 tokens x 8
    __shared__ float hb[16][260];       // hidden: 16 x 256 (padded)
    __shared__ float ob[16][68];        // ffn out tile: 16 x 64
    const int wave = threadIdx.x >> 5;
    const int lane = threadIdx.x & 31;
    const int ln = lane & 15;
    const int lh = lane >> 4;
    const int mtile = blockIdx.x;

    // --- quantum encoder: per-wire factor cos(x)*cos(ry), prefix products ---
    if (threadIdx.x < 16) {
        const int m = mtile * 16 + threadIdx.x;
        float f[8];
#pragma unroll
        for (int j = 0; j < 8; ++j)
            f[j] = __cosf(x1[m * EE + j]) * __cosf(ffn_ry[j]);
        float z0 = 1.f;
#pragma unroll
        for (int j = 1; j < 8; ++j) z0 *= f[j];
        zb[threadIdx.x][0] = fmaxf(z0, 0.f);
        float pref = f[0];
#pragma unroll
        for (int j = 1; j < 8; ++j) {
            pref *= f[j];
            zb[threadIdx.x][j] = fmaxf(pref, 0.f);
        }
    }
    __syncthreads();

    // --- GEMM1: [16x8] x [8x256] -> hidden, 16 col-tiles (4 per wave) ---
#pragma unroll
    for (int t = 0; t < 4; ++t) {
        const int nt = wave * 4 + t;
        v8f hacc = zero8();
#pragma unroll
        for (int s = 0; s < 2; ++s) {
            const int k = s * 4 + lh * 2;
            v2f a = { zb[ln][k], zb[ln][k + 1] };
            const float* wr = w1 + (nt * 16 + ln) * 8 + k;
            v2f b = { wr[0], wr[1] };
            hacc = wmma4(a, b, hacc);
        }
        const int col = nt * 16 + ln;
        const float bias = b1[col];
#pragma unroll
        for (int i = 0; i < 8; ++i)
            hb[i + 8 * lh][col] = hacc[i] + bias;
    }
    __syncthreads();

    // --- GEMM2: [16x256] x [256x64] -> ffn_out tile (one 16-col tile/wave) --
    {
        const int nt = wave;
        v8f oacc = zero8();
#pragma unroll
        for (int s = 0; s < 64; ++s) {
            const int k = s * 4 + lh * 2;
            v2f a = { hb[ln][k], hb[ln][k + 1] };
            const float* wr = w2 + (nt * 16 + ln) * FFN_N + k;
            v2f b = { wr[0], wr[1] };
            oacc = wmma4(a, b, oacc);
        }
        const int col = nt * 16 + ln;
        const float bias = b2[col];
#pragma unroll
        for (int i = 0; i < 8; ++i)
            ob[i + 8 * lh][col] = oacc[i] + bias;
    }
    __syncthreads();

    // --- residual + LayerNorm2 -> out ---
    if (threadIdx.x < 16) {
        const int r = threadIdx.x;
        const int m = mtile * 16 + r;
        float mean = 0.f;
        for (int c = 0; c < EE; ++c)
            mean += ob[r][c] + x1[m * EE + c];
        mean *= (1.f / EE);
        float var = 0.f;
        for (int c = 0; c < EE; ++c) {
            const float d = ob[r][c] + x1[m * EE + c] - mean;
            var += d * d;
        }
        var *= (1.f / EE);
        const float inv = __frsqrt_rn(var + LN_EPS);
        for (int c = 0; c < EE; ++c) {
            const float v = ob[r][c] + x1[m * EE + c];
            out[m * EE + c] = (v - mean) * inv * g[c] + bt[c];
        }
    }
}

// ===========================================================================
extern "C" void kernel_launch(void* const* d_in, const int* in_sizes, int n_in,
                              void* d_out, int out_size, void* d_ws, size_t ws_size,
                              hipStream_t stream) {
    const float* x       = (const float*)d_in[0];
    const float* wq      = (const float*)d_in[1];
    const float* wk      = (const float*)d_in[2];
    const float* wv      = (const float*)d_in[3];
    const float* wo      = (const float*)d_in[4];
    const float* head_rx = (const float*)d_in[5];
    const float* ffn_ry  = (const float*)d_in[6];
    const float* ffn_w1  = (const float*)d_in[7];
    const float* ffn_b1  = (const float*)d_in[8];
    const float* ffn_w2  = (const float*)d_in[9];
    const float* ffn_b2  = (const float*)d_in[10];
    const float* ln1_g   = (const float*)d_in[11];
    const float* ln1_b   = (const float*)d_in[12];
    const float* ln2_g   = (const float*)d_in[13];
    const float* ln2_b   = (const float*)d_in[14];
    float* outp = (float*)d_out;

    float* ws = (float*)d_ws;
    float* qp = ws;                       // B*H*S*DK = 524288
    float* kp = qp + BB * HH * SS * DKK;  // 524288
    float* vp = kp + BB * HH * SS * DKK;  // 524288
    float* ao = vp + BB * HH * SS * DKK;  // B*S*E = 524288
    float* x1 = ao + MTOT * EE;           // 524288

    // 1) QKV projection + per-head quantum circuit
    k_proj_quantum<<<dim3(MTOT / 16, 3), 128, 0, stream>>>(
        x, wq, wk, wv, head_rx, qp, kp, vp);

    // 2) flash attention over S=1024 per (b,h)
    k_flash_attn<<<dim3(SS / 16 / 4, BB * HH), 128, 0, stream>>>(qp, kp, vp, ao);

    // 3) output projection + residual + LN1
    k_outproj_ln1<<<dim3(MTOT / 16), 128, 0, stream>>>(ao, wo, x, ln1_g, ln1_b, x1);

    // 4) quantum FFN + residual + LN2
    k_ffn_ln2<<<dim3(MTOT / 16), 128, 0, stream>>>(
        x1, ffn_ry, ffn_w1, ffn_b1, ffn_w2, ffn_b2, ln2_g, ln2_b, outp);
}